// Conv2d_NN_36378372997762
// MI455X (gfx1250) — compile-verified
//
#include <hip/hip_runtime.h>

// Problem constants from the reference
constexpr int NB   = 8;     // batch
constexpr int CH   = 32;    // channels
constexpr int NPIX = 3136;  // 56*56 tokens
constexpr int KN   = 9;     // neighbors / conv kernel
constexpr int KTOT = CH * KN;       // 288 (flattened conv reduction dim)
constexpr int NT   = NPIX / 16;     // 196 tiles of 16 tokens

typedef __attribute__((ext_vector_type(2))) float v2f;
typedef __attribute__((ext_vector_type(8))) float v8f;

// ---------------------------------------------------------------------------
// Stage A: column L2-normalize; write xnT in token-major [b][n][c] layout so
// WMMA fragment loads are contiguous float2s per lane.
// ---------------------------------------------------------------------------
__global__ void nn_normalize(const float* __restrict__ x, float* __restrict__ xnT) {
    int gid = blockIdx.x * blockDim.x + threadIdx.x;  // b*NPIX + n
    if (gid >= NB * NPIX) return;
    int b = gid / NPIX, n = gid - b * NPIX;
    const float* xb = x + (size_t)b * CH * NPIX + n;
    float v[CH];
    float s = 0.f;
#pragma unroll
    for (int c = 0; c < CH; ++c) { v[c] = xb[(size_t)c * NPIX]; s += v[c] * v[c]; }
    float inv = 1.0f / fmaxf(sqrtf(s), 1e-12f);
    float* o = xnT + (size_t)gid * CH;
#pragma unroll
    for (int c = 0; c < CH; ++c) o[c] = v[c] * inv;
}

// ---------------------------------------------------------------------------
// Stage B: similarity via V_WMMA_F32_16X16X4_F32 + running per-row top-9.
// One wave handles a 16-query tile; 4 independent waves per block.
// ---------------------------------------------------------------------------
__launch_bounds__(128)
__global__ void nn_sim_topk(const float* __restrict__ xnT, int* __restrict__ idxout) {
    __shared__ float tile[4][16][16];
    __shared__ float mbuf[4][32][2 * KN];

    const int lane = threadIdx.x & 31;
    const int wv   = threadIdx.x >> 5;
    const int b    = blockIdx.y;
    const int qtile = blockIdx.x * 4 + wv;         // 0..195
    const int qbase = qtile * 16;

    const float* xb = xnT + (size_t)b * NPIX * CH;

    const int half = lane >> 4;      // 0: rows 0..7 / K 0..1,  1: rows 8..15 / K 2..3
    const int lid  = lane & 15;
    const int coff = half * 2;

    // Preload A fragments (16 queries x 32 channels -> 8 kk steps of K=4)
    v2f afrag[8];
    {
        const float* qp = xb + (size_t)(qbase + lid) * CH;
#pragma unroll
        for (int kk = 0; kk < 8; ++kk) {
            afrag[kk].x = qp[4 * kk + coff];
            afrag[kk].y = qp[4 * kk + coff + 1];
        }
    }

    // Private sorted-descending top-9 (values clipped to [-1,1]; diag = 1.1)
    float tv[KN]; int ti[KN];
#pragma unroll
    for (int j = 0; j < KN; ++j) { tv[j] = -2.0f; ti[j] = 0; }

    const int q = lid;               // query row this lane ranks
    const int qglob = qbase + q;

    for (int t = 0; t < NT; ++t) {
        const int tbase = t * 16;
        const float* cp = xb + (size_t)(tbase + lid) * CH;
        v8f acc = {};
#pragma unroll
        for (int kk = 0; kk < 8; ++kk) {
            v2f bfrag;
            bfrag.x = cp[4 * kk + coff];
            bfrag.y = cp[4 * kk + coff + 1];
            acc = __builtin_amdgcn_wmma_f32_16x16x4_f32(
                false, afrag[kk], false, bfrag, (short)0, acc, false, false);
        }
        // Spill 16x16 tile: lane holds rows r+half*8, column lid
#pragma unroll
        for (int r = 0; r < 8; ++r) tile[wv][r + half * 8][lid] = acc[r];
        asm volatile("s_wait_dscnt 0" ::: "memory");

        // Each lane scans 8 columns of its query row; branchless sorted insert
#pragma unroll
        for (int cc = 0; cc < 8; ++cc) {
            const int col = half * 8 + cc;
            float v = tile[wv][q][col];
            v = fminf(fmaxf(v, -1.0f), 1.0f);
            const int ci = tbase + col;
            if (ci == qglob) v = 1.1f;         // self always best (select, not branch)
            float cv = v; int cidx = ci;
#pragma unroll
            for (int j = 0; j < KN; ++j) {     // strict > keeps lower-index ties
                const bool sw = cv > tv[j];
                const float nv = sw ? tv[j] : cv;
                const int   ni = sw ? ti[j] : cidx;
                tv[j] = sw ? cv   : tv[j];
                ti[j] = sw ? cidx : ti[j];
                cv = nv; cidx = ni;
            }
        }
    }

    // Merge the two half-lane lists per query row (static, branchless)
#pragma unroll
    for (int j = 0; j < KN; ++j) {
        mbuf[wv][lane][j]      = tv[j];
        mbuf[wv][lane][KN + j] = __int_as_float(ti[j]);
    }
    asm volatile("s_wait_dscnt 0" ::: "memory");

    if (half == 0) {
        // Insert partner lane's sorted list into ours; ties -> lower index wins
#pragma unroll
        for (int jj = 0; jj < KN; ++jj) {
            float cv  = mbuf[wv][lane + 16][jj];
            int   cidx = __float_as_int(mbuf[wv][lane + 16][KN + jj]);
#pragma unroll
            for (int j = 0; j < KN; ++j) {
                const bool sw = (cv > tv[j]) || ((cv == tv[j]) && (cidx < ti[j]));
                const float nv = sw ? tv[j] : cv;
                const int   ni = sw ? ti[j] : cidx;
                tv[j] = sw ? cv   : tv[j];
                ti[j] = sw ? cidx : ti[j];
                cv = nv; cidx = ni;
            }
        }
        int* outp = idxout + ((size_t)b * NPIX + qglob) * KN;
#pragma unroll
        for (int k = 0; k < KN; ++k) outp[k] = ti[k];
    }
}

// ---------------------------------------------------------------------------
// Stage C: gather neighbors into LDS, then out[32,16] = W[32,288] x G[288,16]
// via 2 x 72 fp32 WMMA steps. One wave per 16-token tile.
// ---------------------------------------------------------------------------
__launch_bounds__(32)
__global__ void nn_gather_conv(const float* __restrict__ x, const float* __restrict__ w,
                               const float* __restrict__ bias, const int* __restrict__ idx,
                               float* __restrict__ out) {
    __shared__ float G[16][KTOT + 4];   // token-major, padded

    const int lane  = threadIdx.x;      // == input channel during gather
    const int b     = blockIdx.y;
    const int nbase = blockIdx.x * 16;
    const float* xb = x + (size_t)b * CH * NPIX;

    // Gather: G[n][i*9+k] = x[b][i][ idx[b][nbase+n][k] ]
    for (int n = 0; n < 16; ++n) {
        const int* ip = idx + ((size_t)b * NPIX + nbase + n) * KN;
#pragma unroll
        for (int k = 0; k < KN; ++k) {
            int col = ip[k];                       // uniform across lanes
            G[n][lane * KN + k] = xb[(size_t)lane * NPIX + col];
        }
    }
    asm volatile("s_wait_dscnt 0" ::: "memory");

    const int half = lane >> 4;
    const int lid  = lane & 15;
    const int coff = half * 2;

    for (int mt = 0; mt < 2; ++mt) {
        const int o0 = mt * 16;
        v8f acc;
#pragma unroll
        for (int r = 0; r < 8; ++r) acc[r] = bias[o0 + r + half * 8];

        const float* wrow = w + (size_t)(o0 + lid) * KTOT;   // A row for this lane
        for (int kk = 0; kk < KTOT / 4; ++kk) {              // 72 K-steps
            const int kf = 4 * kk + coff;
            v2f a;  a.x  = wrow[kf];     a.y  = wrow[kf + 1];
            v2f bf; bf.x = G[lid][kf];   bf.y = G[lid][kf + 1];
            acc = __builtin_amdgcn_wmma_f32_16x16x4_f32(
                false, a, false, bf, (short)0, acc, false, false);
        }
        float* op = out + ((size_t)b * CH + o0) * NPIX + nbase + lid;
#pragma unroll
        for (int r = 0; r < 8; ++r) op[(size_t)(r + half * 8) * NPIX] = acc[r];
    }
}

// ---------------------------------------------------------------------------
extern "C" void kernel_launch(void* const* d_in, const int* in_sizes, int n_in,
                              void* d_out, int out_size, void* d_ws, size_t ws_size,
                              hipStream_t stream) {
    const float* x    = (const float*)d_in[0];   // [8,32,56,56]
    const float* w    = (const float*)d_in[1];   // [32,32,9]
    const float* bias = (const float*)d_in[2];   // [32]
    float* out = (float*)d_out;                  // [8,32,56,56]

    float* xnT = (float*)d_ws;                                   // 8*3136*32 f32
    int*   idx = (int*)((char*)d_ws +
                        (size_t)NB * NPIX * CH * sizeof(float)); // 8*3136*9 i32

    nn_normalize<<<(NB * NPIX + 255) / 256, 256, 0, stream>>>(x, xnT);

    dim3 gB(NT / 4, NB);                 // 49 x 8 blocks, 4 waves each
    nn_sim_topk<<<gB, 128, 0, stream>>>(xnT, idx);

    dim3 gC(NT, NB);                     // 196 x 8 blocks, 1 wave each
    nn_gather_conv<<<gC, 32, 0, stream>>>(x, w, bias, idx, out);
}